// Mamba_38482906972962
// MI455X (gfx1250) — compile-verified
//
#include <hip/hip_runtime.h>

#define D_MODEL 4096
#define D_STATE 16
#define D_CONV  4
#define D_INNER 8192
#define DT_RANK 256

typedef float v2f __attribute__((ext_vector_type(2)));
typedef float v8f __attribute__((ext_vector_type(8)));

__device__ __forceinline__ float sigmoid_fast(float v) {
  return 1.0f / (1.0f + __expf(-v));
}

// ---------------------------------------------------------------------------
// WMMA-f32 GEMV: out[j] = act( sum_k W[j][k] * x[k] (+ bias[j]) )
// One wave32 computes 16 consecutive outputs with V_WMMA_F32_16X16X4_F32.
//   A (16x4, MxK): x-chunk replicated across M rows
//       lane layout: lanes 0-15 -> K={0,1}; lanes 16-31 -> K={2,3}
//   B (4x16, KxN): weight tile; VGPR v, lanes 0-15: K=v, N=lane;
//       lanes 16-31: K=2+v, N=lane-16
//   D: all M rows identical; lane n<16 holds result for output j0+n in c[0].
// ACT: 0 = identity, 1 = softplus(v + bias)
// ---------------------------------------------------------------------------
template <int ACT>
__global__ __launch_bounds__(256) void gemv_wmma_f32(
    const float* __restrict__ W, const float* __restrict__ x,
    const float* __restrict__ bias, float* __restrict__ out,
    int n_out, int K) {
  const int lane = threadIdx.x & 31;
  const int wave = (int)((blockIdx.x * blockDim.x + threadIdx.x) >> 5);
  const int j0 = wave * 16;
  if (j0 >= n_out) return;                    // wave-uniform: EXEC all-1s below

  const int half = lane >> 4;                 // 0 -> K {0,1}, 1 -> K {2,3}
  const int n    = lane & 15;
  const float* __restrict__ wrow = W + (size_t)(j0 + n) * (size_t)K + 2 * half;
  const float* __restrict__ xp   = x + 2 * half;

  v8f acc0 = {0.f, 0.f, 0.f, 0.f, 0.f, 0.f, 0.f, 0.f};
  v8f acc1 = acc0, acc2 = acc0, acc3 = acc0;

  // 4 independent accumulators break the WMMA C->C dependency chain.
  for (int k = 0; k < K; k += 16) {
    v2f a0 = *(const v2f*)(xp + k);
    v2f b0 = *(const v2f*)(wrow + k);
    v2f a1 = *(const v2f*)(xp + k + 4);
    v2f b1 = *(const v2f*)(wrow + k + 4);
    v2f a2 = *(const v2f*)(xp + k + 8);
    v2f b2 = *(const v2f*)(wrow + k + 8);
    v2f a3 = *(const v2f*)(xp + k + 12);
    v2f b3 = *(const v2f*)(wrow + k + 12);
    acc0 = __builtin_amdgcn_wmma_f32_16x16x4_f32(false, a0, false, b0,
                                                 (short)0, acc0, false, false);
    acc1 = __builtin_amdgcn_wmma_f32_16x16x4_f32(false, a1, false, b1,
                                                 (short)0, acc1, false, false);
    acc2 = __builtin_amdgcn_wmma_f32_16x16x4_f32(false, a2, false, b2,
                                                 (short)0, acc2, false, false);
    acc3 = __builtin_amdgcn_wmma_f32_16x16x4_f32(false, a3, false, b3,
                                                 (short)0, acc3, false, false);
  }
  v8f ct = (acc0 + acc1) + (acc2 + acc3);

  if (lane < 16) {
    float r = ct[0];
    const int j = j0 + lane;
    if (ACT == 1) {
      r += bias[j];
      r = (r > 20.0f) ? r : __logf(1.0f + __expf(r));   // stable softplus
    }
    out[j] = r;
  }
}

// ---------------------------------------------------------------------------
// Depthwise causal conv step + SiLU. One thread per channel (8192).
// conv_state_new[c][0..2] = conv_state[c][1..3]; [3] = x[c]
// x_conv[c] = silu(dot(conv_state_new[c], conv_w[c]) + conv_b[c])
// ---------------------------------------------------------------------------
__global__ __launch_bounds__(256) void conv_step_kernel(
    const float* __restrict__ conv_state, const float* __restrict__ x,
    const float* __restrict__ conv_w, const float* __restrict__ conv_b,
    float* __restrict__ conv_state_new, float* __restrict__ x_conv) {
  const int c = blockIdx.x * blockDim.x + threadIdx.x;
  if (c >= D_INNER) return;
  float4 s = ((const float4*)conv_state)[c];
  float4 w = ((const float4*)conv_w)[c];
  float4 ns;
  ns.x = s.y; ns.y = s.z; ns.z = s.w; ns.w = x[c];
  float acc = ns.x * w.x + ns.y * w.y + ns.z * w.z + ns.w * w.w + conv_b[c];
  ((float4*)conv_state_new)[c] = ns;
  x_conv[c] = acc * sigmoid_fast(acc);
}

// ---------------------------------------------------------------------------
// Selective-SSM state update + output gating. One thread per channel (8192).
// dA = exp(-dt * exp(A_log));  s' = s*dA + (x*dt)*B;  y = s'@C + D*x; y *= silu(z)
// ---------------------------------------------------------------------------
__global__ __launch_bounds__(256) void ssm_step_kernel(
    const float* __restrict__ ssm_state, const float* __restrict__ A_log,
    const float* __restrict__ x_conv, const float* __restrict__ dt,
    const float* __restrict__ x_db,    // [dt_raw(256) | B(16) | C(16)]
    const float* __restrict__ Dvec, const float* __restrict__ z,
    float* __restrict__ ssm_state_new, float* __restrict__ y) {
  const int c = blockIdx.x * blockDim.x + threadIdx.x;
  if (c >= D_INNER) return;
  const float* __restrict__ Bv = x_db + DT_RANK;
  const float* __restrict__ Cv = x_db + DT_RANK + D_STATE;
  const float dtc = dt[c];
  const float xc  = x_conv[c];
  const float xdt = xc * dtc;
  float acc = 0.0f;
  const float4* Sr = (const float4*)(ssm_state + (size_t)c * D_STATE);
  const float4* Ar = (const float4*)(A_log + (size_t)c * D_STATE);
  float4* Nr = (float4*)(ssm_state_new + (size_t)c * D_STATE);
#pragma unroll
  for (int q = 0; q < 4; ++q) {
    float4 s  = Sr[q];
    float4 al = Ar[q];
    float4 ns;
    ns.x = s.x * __expf(-__expf(al.x) * dtc) + xdt * Bv[4 * q + 0];
    ns.y = s.y * __expf(-__expf(al.y) * dtc) + xdt * Bv[4 * q + 1];
    ns.z = s.z * __expf(-__expf(al.z) * dtc) + xdt * Bv[4 * q + 2];
    ns.w = s.w * __expf(-__expf(al.w) * dtc) + xdt * Bv[4 * q + 3];
    acc += ns.x * Cv[4 * q + 0] + ns.y * Cv[4 * q + 1] +
           ns.z * Cv[4 * q + 2] + ns.w * Cv[4 * q + 3];
    Nr[q] = ns;
  }
  float yv = acc + Dvec[c] * xc;
  const float zc = z[c];
  yv *= zc * sigmoid_fast(zc);
  y[c] = yv;
}

// ---------------------------------------------------------------------------
extern "C" void kernel_launch(void* const* d_in, const int* in_sizes, int n_in,
                              void* d_out, int out_size, void* d_ws, size_t ws_size,
                              hipStream_t stream) {
  (void)in_sizes; (void)n_in; (void)out_size; (void)ws_size;
  const float* hidden     = (const float*)d_in[0];   // 4096
  const float* conv_state = (const float*)d_in[1];   // 8192*4
  const float* ssm_state  = (const float*)d_in[2];   // 8192*16
  const float* in_proj_w  = (const float*)d_in[3];   // 16384*4096
  const float* conv_w     = (const float*)d_in[4];   // 8192*4
  const float* conv_b     = (const float*)d_in[5];   // 8192
  const float* x_proj_w   = (const float*)d_in[6];   // 288*8192
  const float* dt_proj_w  = (const float*)d_in[7];   // 8192*256
  const float* dt_proj_b  = (const float*)d_in[8];   // 8192
  const float* A_log      = (const float*)d_in[9];   // 8192*16
  const float* Dvec       = (const float*)d_in[10];  // 8192
  const float* out_proj_w = (const float*)d_in[11];  // 4096*8192

  // Outputs concatenated flat: out (4096), conv_state_new (32768), ssm_state_new (131072)
  float* out_y          = (float*)d_out;
  float* conv_state_new = out_y + D_MODEL;
  float* ssm_state_new  = conv_state_new + D_INNER * D_CONV;

  // Workspace layout (floats)
  float* ws     = (float*)d_ws;
  float* xz     = ws;                    // 16384  (x | z)
  float* xvec   = xz;                    // first 8192
  float* zvec   = xz + D_INNER;          // second 8192
  float* x_conv = ws + 2 * D_INNER;      // 8192
  float* x_db   = x_conv + D_INNER;      // 288 (padded to 512)
  float* dtv    = x_db + 512;            // 8192
  float* yvec   = dtv + D_INNER;         // 8192

  const int BLK = 256;                   // 8 wave32 per block

  // 1) xz = in_proj_w @ h              (16384 outputs, K=4096)
  {
    int tiles = (2 * D_INNER) / 16;
    gemv_wmma_f32<0><<<(tiles + 7) / 8, BLK, 0, stream>>>(
        in_proj_w, hidden, nullptr, xz, 2 * D_INNER, D_MODEL);
  }
  // 2) conv shift + depthwise conv + SiLU
  conv_step_kernel<<<D_INNER / BLK, BLK, 0, stream>>>(
      conv_state, xvec, conv_w, conv_b, conv_state_new, x_conv);
  // 3) x_db = x_proj_w @ x_conv        (288 outputs, K=8192)
  {
    int tiles = (DT_RANK + 2 * D_STATE) / 16;
    gemv_wmma_f32<0><<<(tiles + 7) / 8, BLK, 0, stream>>>(
        x_proj_w, x_conv, nullptr, x_db, DT_RANK + 2 * D_STATE, D_INNER);
  }
  // 4) dt = softplus(dt_proj_w @ dt_raw + dt_proj_b)   (8192 outputs, K=256)
  {
    int tiles = D_INNER / 16;
    gemv_wmma_f32<1><<<(tiles + 7) / 8, BLK, 0, stream>>>(
        dt_proj_w, x_db, dt_proj_b, dtv, D_INNER, DT_RANK);
  }
  // 5) SSM state update + gating
  ssm_step_kernel<<<D_INNER / BLK, BLK, 0, stream>>>(
      ssm_state, A_log, x_conv, dtv, x_db, Dvec, zvec, ssm_state_new, yvec);
  // 6) out = out_proj_w @ y            (4096 outputs, K=8192)
  {
    int tiles = D_MODEL / 16;
    gemv_wmma_f32<0><<<(tiles + 7) / 8, BLK, 0, stream>>>(
        out_proj_w, yvec, nullptr, out_y, D_MODEL, D_INNER);
  }
}